// GAT_27109833572874
// MI455X (gfx1250) — compile-verified
//
#include <hip/hip_runtime.h>
#include <math.h>

// ---- problem dims (fixed by reference) ----
constexpr int kN  = 4096;
constexpr int kF  = 512;
constexpr int kH  = 8;
constexpr int kFh = 64;
constexpr int kC  = 40;
constexpr int kCp = 48;            // C padded to 3 WMMA col-tiles
constexpr float kAlpha  = 0.2f;
constexpr float kNegInf = -9.0e15f;

typedef __attribute__((ext_vector_type(2))) float v2f;
typedef __attribute__((ext_vector_type(8))) float v8f;
typedef __attribute__((ext_vector_type(4))) unsigned int u32x4;
typedef __attribute__((ext_vector_type(8))) int i32x8;
typedef __attribute__((ext_vector_type(4))) int i32x4;

__device__ __forceinline__ float lrelu(float v) { return v > 0.f ? v : kAlpha * v; }
__device__ __forceinline__ float elu_f(float v) { return v > 0.f ? v : (__expf(v) - 1.f); }

// D = A(16x4 f32) * B(4x16 f32) + C(16x16 f32)  -> V_WMMA_F32_16X16X4_F32
__device__ __forceinline__ v8f wmma4(v2f a, v2f b, v8f c) {
  return __builtin_amdgcn_wmma_f32_16x16x4_f32(false, a, false, b, (short)0, c, false, false);
}

// Flat shared-aperture addresses keep the LDS address in bits [31:0] (ISA 10.2).
__device__ __forceinline__ unsigned lds_addr_of(const void* p) {
  return (unsigned)(uintptr_t)p;
}

// TDM: contiguous 1-D copy of `nfloats` f32 from global `src` into LDS at
// byte address `lds_bytes`. D# per ISA 8.3/8.4: count=1, type=2 ("image"),
// data_size=2 (4B), tile_dim0=nfloats, tensor_dim0=nfloats, no pad/iterate.
// This toolchain exposes the 6-arg builtin: (g0, g1, g2, g3, g?, cpol).
__device__ __forceinline__ void tdm_load_f32(const float* src, unsigned lds_bytes,
                                             int nfloats) {
  unsigned long long ga = (unsigned long long)(uintptr_t)src;
  u32x4 g0;
  g0[0] = 1u;                                              // count=1, user D#
  g0[1] = lds_bytes;                                       // lds_addr
  g0[2] = (unsigned)(ga & 0xFFFFFFFFu);                    // global_addr[31:0]
  g0[3] = (unsigned)((ga >> 32) & 0x1FFFFFFu) | (2u << 30); // addr[56:32] | type=2
  i32x8 g1;
  g1[0] = 0x00020000;                                      // data_size = 4B
  g1[1] = (nfloats & 0xFFFF) << 16;                        // tensor_dim0[15:0]
  g1[2] = ((nfloats >> 16) & 0xFFFF) | (1 << 16);          // tensor_dim0[31:16] | tensor_dim1=1
  g1[3] = (nfloats & 0xFFFF) << 16;                        // tile_dim0
  g1[4] = 0;                                               // tile_dim1/2 unused
  g1[5] = nfloats;                                         // tensor_dim0_stride
  g1[6] = 0;
  g1[7] = 0;
  i32x4 z4 = {0, 0, 0, 0};
  i32x8 z8 = {0, 0, 0, 0, 0, 0, 0, 0};
  __builtin_amdgcn_tensor_load_to_lds(g0, g1, z4, z4, z8, 0);
}

// ---------------------------------------------------------------------------
// K1: Wh1[h] = x @ W1[h]; 8 waves/block share head h, W1[h] k-chunks staged
// into LDS by TDM (double buffered). Fused f_src/f_dst epilogue.
// blockIdx: head = b>>5, rowblocks (b&31)*8 + wave
// ---------------------------------------------------------------------------
__global__ void k1_gemm_xw1(const float* __restrict__ x, const float* __restrict__ W1,
                            const float* __restrict__ a1,
                            float* __restrict__ Wh1, float* __restrict__ fsrc1,
                            float* __restrict__ fdst1T) {
  __shared__ float sB[2][64 * kFh];                // 2 x 16KB
  const int waveIn = threadIdx.x >> 5;             // 0..7
  const int lane = threadIdx.x & 31;
  const int head = blockIdx.x >> 5;
  const int rowBase = ((blockIdx.x & 31) * 8 + waveIn) * 16;
  const int l15 = lane & 15;
  const int hi  = lane >> 4;

  const float* W = W1 + (size_t)head * kF * kFh;
  const int arow = rowBase + l15;

  v8f acc[4] = {};
  // prologue: chunk 0
  tdm_load_f32(W + (size_t)(waveIn * 8) * kFh,
               lds_addr_of(&sB[0][0]) + waveIn * 8 * kFh * 4, 8 * kFh);
  constexpr int NCH = kF / 64;                     // 8 chunks
  for (int ch = 0; ch < NCH; ++ch) {
    const int k0 = ch * 64;
    if (ch + 1 < NCH) {
      tdm_load_f32(W + (size_t)(k0 + 64 + waveIn * 8) * kFh,
                   lds_addr_of(&sB[(ch + 1) & 1][0]) + waveIn * 8 * kFh * 4, 8 * kFh);
      __builtin_amdgcn_s_wait_tensorcnt(1);
    } else {
      __builtin_amdgcn_s_wait_tensorcnt(0);
    }
    __syncthreads();
    const float* sb = &sB[ch & 1][0];
    for (int kk = 0; kk < 64; kk += 4) {
      const int lk = kk + 2 * hi;
      v2f A;
      A.x = x[arow * kF + k0 + lk];
      A.y = x[arow * kF + k0 + lk + 1];
#pragma unroll
      for (int t = 0; t < 4; ++t) {
        const int col = 16 * t + l15;
        v2f B;
        B.x = sb[lk * kFh + col];
        B.y = sb[(lk + 1) * kFh + col];
        acc[t] = wmma4(A, B, acc[t]);
      }
    }
    __syncthreads();
  }

  float* WhOut = Wh1 + (size_t)head * kN * kFh;
  const float* av = a1 + head * 2 * kFh;           // [0..63] src, [64..127] dst
#pragma unroll
  for (int r = 0; r < 8; ++r) {
    const int row = rowBase + r + 8 * hi;
    float ss = 0.f, sd = 0.f;
#pragma unroll
    for (int t = 0; t < 4; ++t) {
      const int col = 16 * t + l15;
      const float v = acc[t][r];
      WhOut[row * kFh + col] = v;
      ss += v * av[col];
      sd += v * av[kFh + col];
    }
#pragma unroll
    for (int m = 1; m < 16; m <<= 1) {             // reduce within 16-lane halves
      ss += __shfl_xor(ss, m, 32);
      sd += __shfl_xor(sd, m, 32);
    }
    if (l15 == 0) {
      fsrc1[head * kN + row] = ss;
      fdst1T[row * kH + head] = sd;                // transposed for per-j access
    }
  }
}

// ---------------------------------------------------------------------------
// K2: one wave per row: m1[h][i] = LeakyReLU(f_src[h][i] + max_{adj} f_dst[h][j])
// (LeakyReLU monotonic => exact softmax row-max), all 8 heads in one adj pass
// ---------------------------------------------------------------------------
__global__ void k2_rowmax1(const int* __restrict__ adj, const float* __restrict__ fdst1T,
                           const float* __restrict__ fsrc1, float* __restrict__ m1) {
  const int wave = (blockIdx.x * blockDim.x + threadIdx.x) >> 5;
  const int lane = threadIdx.x & 31;
  const int row = wave;
  const int* arow = adj + (size_t)row * kN;

  float mx[kH];
#pragma unroll
  for (int h = 0; h < kH; ++h) mx[h] = -3.0e38f;

  for (int j = lane; j < kN; j += 32) {
    if (arow[j] > 0) {
      const float* fd = fdst1T + j * kH;
#pragma unroll
      for (int h = 0; h < kH; ++h) mx[h] = fmaxf(mx[h], fd[h]);
    }
  }
#pragma unroll
  for (int h = 0; h < kH; ++h)
#pragma unroll
    for (int m = 1; m < 32; m <<= 1) mx[h] = fmaxf(mx[h], __shfl_xor(mx[h], m, 32));

  if (lane == 0) {
#pragma unroll
    for (int h = 0; h < kH; ++h)
      m1[h * kN + row] = (mx[h] > -1.0e38f) ? lrelu(fsrc1[h * kN + row] + mx[h]) : kNegInf;
  }
}

// ---------------------------------------------------------------------------
// K3: flash-style fused masked-softmax @ Wh1 per head; ELU; concat into hcat.
// Scores are rank-1 (f_src[i] + f_dst[j]) -> no QK^T GEMM. 8 waves/block share
// head; Wh1 j-chunks staged into LDS via TDM double-buffering.
// ---------------------------------------------------------------------------
__global__ void k3_attn1(const int* __restrict__ adj, const float* __restrict__ Wh1,
                         const float* __restrict__ fsrc1, const float* __restrict__ fdst1T,
                         const float* __restrict__ m1, float* __restrict__ hcat) {
  __shared__ float sB[2][64 * kFh];                // 2 x 16KB
  const int waveIn = threadIdx.x >> 5;
  const int lane = threadIdx.x & 31;
  const int head = blockIdx.x >> 5;
  const int rowBase = ((blockIdx.x & 31) * 8 + waveIn) * 16;
  const int l15 = lane & 15;
  const int hi  = lane >> 4;

  const int myrow = rowBase + l15;
  const float fs = fsrc1[head * kN + myrow];
  const float mm = m1[head * kN + myrow];
  const float* Wh = Wh1 + (size_t)head * kN * kFh;
  const int* arow = adj + (size_t)myrow * kN;

  v8f acc[4] = {};
  float lsum = 0.f;

  tdm_load_f32(Wh + (size_t)(waveIn * 8) * kFh,
               lds_addr_of(&sB[0][0]) + waveIn * 8 * kFh * 4, 8 * kFh);
  constexpr int NCH = kN / 64;                     // 64 chunks
  for (int ch = 0; ch < NCH; ++ch) {
    const int j0 = ch * 64;
    if (ch + 1 < NCH) {
      tdm_load_f32(Wh + (size_t)(j0 + 64 + waveIn * 8) * kFh,
                   lds_addr_of(&sB[(ch + 1) & 1][0]) + waveIn * 8 * kFh * 4, 8 * kFh);
      __builtin_amdgcn_s_wait_tensorcnt(1);
    } else {
      __builtin_amdgcn_s_wait_tensorcnt(0);
    }
    __syncthreads();
    const float* sb = &sB[ch & 1][0];
    for (int kk = 0; kk < 64; kk += 4) {
      const int lk = kk + 2 * hi;
      const int ja = j0 + lk;
      const int a0 = arow[ja];
      const int a1v = arow[ja + 1];
      const float fd0 = fdst1T[ja * kH + head];
      const float fd1 = fdst1T[(ja + 1) * kH + head];
      const float p0 = (a0 > 0) ? __expf(lrelu(fs + fd0) - mm) : 0.f;
      const float p1 = (a1v > 0) ? __expf(lrelu(fs + fd1) - mm) : 0.f;
      lsum += p0 + p1;
      v2f A; A.x = p0; A.y = p1;
#pragma unroll
      for (int t = 0; t < 4; ++t) {
        const int col = 16 * t + l15;
        v2f B;
        B.x = sb[lk * kFh + col];
        B.y = sb[(lk + 1) * kFh + col];
        acc[t] = wmma4(A, B, acc[t]);
      }
    }
    __syncthreads();
  }
  lsum += __shfl_xor(lsum, 16, 32);                // lanes L, L+16 share a row

#pragma unroll
  for (int r = 0; r < 8; ++r) {
    const float lr = __shfl(lsum, r + 8 * hi, 32); // ds_bpermute broadcast
    const float invl = 1.f / fmaxf(lr, 1e-30f);
    const int row = rowBase + r + 8 * hi;
#pragma unroll
    for (int t = 0; t < 4; ++t) {
      const float v = elu_f(acc[t][r] * invl);
      hcat[row * (kH * kFh) + head * kFh + 16 * t + l15] = v;
    }
  }
}

// ---------------------------------------------------------------------------
// K4: Wh2 = hcat @ W2 (cols padded 40->48), fused f_src2/f_dst2 epilogue.
// W2 is only 80KB and shared by all waves -> direct global loads stay in L0/L2.
// ---------------------------------------------------------------------------
__global__ void k4_gemm_hw2(const float* __restrict__ hcat, const float* __restrict__ W2,
                            const float* __restrict__ a2,
                            float* __restrict__ Wh2p, float* __restrict__ fsrc2,
                            float* __restrict__ fdst2) {
  const int wave = (blockIdx.x * blockDim.x + threadIdx.x) >> 5;
  const int lane = threadIdx.x & 31;
  const int rowBase = wave * 16;
  const int l15 = lane & 15;
  const int hi  = lane >> 4;
  const int arow = rowBase + l15;

  v8f acc[3] = {};
  for (int k0 = 0; k0 < kF; k0 += 4) {
    const int ka = k0 + 2 * hi;
    v2f A;
    A.x = hcat[arow * kF + ka];
    A.y = hcat[arow * kF + ka + 1];
#pragma unroll
    for (int t = 0; t < 3; ++t) {
      const int col = 16 * t + l15;
      v2f B;
      B.x = (col < kC) ? W2[ka * kC + col] : 0.f;
      B.y = (col < kC) ? W2[(ka + 1) * kC + col] : 0.f;
      acc[t] = wmma4(A, B, acc[t]);
    }
  }
#pragma unroll
  for (int r = 0; r < 8; ++r) {
    const int row = rowBase + r + 8 * hi;
    float ss = 0.f, sd = 0.f;
#pragma unroll
    for (int t = 0; t < 3; ++t) {
      const int col = 16 * t + l15;
      const float v = acc[t][r];
      Wh2p[row * kCp + col] = (col < kC) ? v : 0.f;
      if (col < kC) { ss += v * a2[col]; sd += v * a2[kC + col]; }
    }
#pragma unroll
    for (int m = 1; m < 16; m <<= 1) {
      ss += __shfl_xor(ss, m, 32);
      sd += __shfl_xor(sd, m, 32);
    }
    if (l15 == 0) { fsrc2[row] = ss; fdst2[row] = sd; }
  }
}

// ---------------------------------------------------------------------------
// K5: layer-2 masked row-max
// ---------------------------------------------------------------------------
__global__ void k5_rowmax2(const int* __restrict__ adj, const float* __restrict__ fdst2,
                           const float* __restrict__ fsrc2, float* __restrict__ m2) {
  const int wave = (blockIdx.x * blockDim.x + threadIdx.x) >> 5;
  const int lane = threadIdx.x & 31;
  const int row = wave;
  const int* arow = adj + (size_t)row * kN;
  float mx = -3.0e38f;
  for (int j = lane; j < kN; j += 32)
    if (arow[j] > 0) mx = fmaxf(mx, fdst2[j]);
#pragma unroll
  for (int m = 1; m < 32; m <<= 1) mx = fmaxf(mx, __shfl_xor(mx, m, 32));
  if (lane == 0) m2[row] = (mx > -1.0e38f) ? lrelu(fsrc2[row] + mx) : kNegInf;
}

// ---------------------------------------------------------------------------
// K6: fused layer-2 attention + ELU + log_softmax -> d_out [N, C].
// Wh2p j-chunks (64 x 48 f32 = 12KB) staged into LDS via TDM double-buffering.
// ---------------------------------------------------------------------------
__global__ void k6_attn2(const int* __restrict__ adj, const float* __restrict__ Wh2p,
                         const float* __restrict__ fsrc2, const float* __restrict__ fdst2,
                         const float* __restrict__ m2, float* __restrict__ out) {
  __shared__ float sB[2][64 * kCp];                // 2 x 12KB
  const int waveIn = threadIdx.x >> 5;
  const int lane = threadIdx.x & 31;
  const int rowBase = (blockIdx.x * 8 + waveIn) * 16;
  const int l15 = lane & 15;
  const int hi  = lane >> 4;

  const int myrow = rowBase + l15;
  const float fs = fsrc2[myrow];
  const float mm = m2[myrow];
  const int* arow = adj + (size_t)myrow * kN;

  v8f acc[3] = {};
  float lsum = 0.f;

  tdm_load_f32(Wh2p + (size_t)(waveIn * 8) * kCp,
               lds_addr_of(&sB[0][0]) + waveIn * 8 * kCp * 4, 8 * kCp);
  constexpr int NCH = kN / 64;
  for (int ch = 0; ch < NCH; ++ch) {
    const int j0 = ch * 64;
    if (ch + 1 < NCH) {
      tdm_load_f32(Wh2p + (size_t)(j0 + 64 + waveIn * 8) * kCp,
                   lds_addr_of(&sB[(ch + 1) & 1][0]) + waveIn * 8 * kCp * 4, 8 * kCp);
      __builtin_amdgcn_s_wait_tensorcnt(1);
    } else {
      __builtin_amdgcn_s_wait_tensorcnt(0);
    }
    __syncthreads();
    const float* sb = &sB[ch & 1][0];
    for (int kk = 0; kk < 64; kk += 4) {
      const int lk = kk + 2 * hi;
      const int ja = j0 + lk;
      const int a0 = arow[ja];
      const int a1v = arow[ja + 1];
      const float p0 = (a0 > 0) ? __expf(lrelu(fs + fdst2[ja]) - mm) : 0.f;
      const float p1 = (a1v > 0) ? __expf(lrelu(fs + fdst2[ja + 1]) - mm) : 0.f;
      lsum += p0 + p1;
      v2f A; A.x = p0; A.y = p1;
#pragma unroll
      for (int t = 0; t < 3; ++t) {
        const int col = 16 * t + l15;
        v2f B;
        B.x = sb[lk * kCp + col];
        B.y = sb[(lk + 1) * kCp + col];
        acc[t] = wmma4(A, B, acc[t]);
      }
    }
    __syncthreads();
  }
  lsum += __shfl_xor(lsum, 16, 32);

#pragma unroll
  for (int r = 0; r < 8; ++r) {
    const float lr = __shfl(lsum, r + 8 * hi, 32);
    const float invl = 1.f / fmaxf(lr, 1e-30f);
    const int row = rowBase + r + 8 * hi;
    float e[3];
    float mx = -3.0e38f;
#pragma unroll
    for (int t = 0; t < 3; ++t) {
      const int col = 16 * t + l15;
      e[t] = elu_f(acc[t][r] * invl);
      if (col < kC) mx = fmaxf(mx, e[t]);
    }
#pragma unroll
    for (int m = 1; m < 16; m <<= 1) mx = fmaxf(mx, __shfl_xor(mx, m, 32));
    float s = 0.f;
#pragma unroll
    for (int t = 0; t < 3; ++t) {
      const int col = 16 * t + l15;
      if (col < kC) s += __expf(e[t] - mx);
    }
#pragma unroll
    for (int m = 1; m < 16; m <<= 1) s += __shfl_xor(s, m, 32);
    const float lse = mx + __logf(s);
#pragma unroll
    for (int t = 0; t < 3; ++t) {
      const int col = 16 * t + l15;
      if (col < kC) out[row * kC + col] = e[t] - lse;
    }
  }
}

// ---------------------------------------------------------------------------
extern "C" void kernel_launch(void* const* d_in, const int* in_sizes, int n_in,
                              void* d_out, int out_size, void* d_ws, size_t ws_size,
                              hipStream_t stream) {
  const float* x   = (const float*)d_in[0];   // [4096,512]
  const int*   adj = (const int*)  d_in[1];   // [4096,4096]
  const float* W1  = (const float*)d_in[2];   // [8,512,64]
  const float* a1  = (const float*)d_in[3];   // [8,128]
  const float* W2  = (const float*)d_in[4];   // [512,40]
  const float* a2  = (const float*)d_in[5];   // [80]
  float* out = (float*)d_out;                 // [4096,40]

  float* ws = (float*)d_ws;
  float* Wh1    = ws;                          // 8*4096*64
  float* fsrc1  = Wh1   + (size_t)kH * kN * kFh;
  float* fdst1T = fsrc1 + (size_t)kH * kN;
  float* m1     = fdst1T+ (size_t)kN * kH;
  float* hcat   = m1    + (size_t)kH * kN;     // 4096*512
  float* Wh2p   = hcat  + (size_t)kN * kH * kFh;
  float* fsrc2  = Wh2p  + (size_t)kN * kCp;
  float* fdst2  = fsrc2 + kN;
  float* m2     = fdst2 + kN;

  k1_gemm_xw1<<<dim3(256), dim3(256), 0, stream>>>(x, W1, a1, Wh1, fsrc1, fdst1T);
  k2_rowmax1 <<<dim3(512), dim3(256), 0, stream>>>(adj, fdst1T, fsrc1, m1);
  k3_attn1   <<<dim3(256), dim3(256), 0, stream>>>(adj, Wh1, fsrc1, fdst1T, m1, hcat);
  k4_gemm_hw2<<<dim3(32),  dim3(256), 0, stream>>>(hcat, W2, a2, Wh2p, fsrc2, fdst2);
  k5_rowmax2 <<<dim3(512), dim3(256), 0, stream>>>(adj, fdst2, fsrc2, m2);
  k6_attn2   <<<dim3(32),  dim3(256), 0, stream>>>(adj, Wh2p, fsrc2, fdst2, m2, out);
}